// SparseFCN_7060926235182
// MI455X (gfx1250) — compile-verified
//
#include <hip/hip_runtime.h>
#include <hip/hip_bf16.h>
#include <stdint.h>

// ---------------------------------------------------------------------------
// SparseFCN on MI455X (gfx1250).
// Key identity: stride == dilation == s  =>  every conv is a dense stride-1
// 3x3 VALID conv on the s-subsampled grid; every conv_adj is a stride-1 full
// correlation on that grid scattered back to sites that are multiples of s.
// The heavy op (final 64->16 1x1 conv over 722k pixels) runs on the matrix
// pipe via V_WMMA_F32_16X16X4_F32 (wave32, 16 pixels x 16 outputs, K=64).
// ---------------------------------------------------------------------------

typedef float v2f __attribute__((ext_vector_type(2)));
typedef float v8f __attribute__((ext_vector_type(8)));

#define H0 601
#define NPIX (2 * H0 * H0)   // 722402 flat (b,h,w) pixels

// ---------------------------------------------------------------------------
// k1: m1 = dilate3x3(m0 subsampled);  y1 = relu((conv3x3(X_sub,w1)+b1)*m1)
// One thread per (b, oy, ox) on the H1 grid; 8 output channels.
// mode 0: Cin=1  from xraw*m0           (block 0)
// mode 1: Cin=17 = 16 from Xmain + xraw*m0 (block 1: concat(x1, xin))
// mode 2: Cin=16 from Xmain             (blocks 2,3)
// ---------------------------------------------------------------------------
__global__ void k1_conv1(const float* __restrict__ Xmain,
                         const float* __restrict__ xraw,
                         const uint8_t* __restrict__ m0,
                         const float* __restrict__ w1,
                         const float* __restrict__ b1,
                         float* __restrict__ y1,
                         uint8_t* __restrict__ m1,
                         int s, int H1, int cin, int mode)
{
    int tid = blockIdx.x * blockDim.x + threadIdx.x;
    int total = 2 * H1 * H1;
    if (tid >= total) return;
    int b  = tid / (H1 * H1);
    int r  = tid - b * H1 * H1;
    int oy = r / H1, ox = r - (r / H1) * H1;

    float acc[8];
#pragma unroll
    for (int c = 0; c < 8; ++c) acc[c] = 0.f;
    bool any = false;

    for (int ky = 0; ky < 3; ++ky) {
        for (int kx = 0; kx < 3; ++kx) {
            int py = s * (oy + ky), px = s * (ox + kx);      // < 601 by construction
            size_t p = ((size_t)b * H0 + py) * H0 + px;
            bool mm = (m0[p] != 0);
            any = any || mm;
            const float* wp = w1 + (size_t)(ky * 3 + kx) * cin * 8;
            for (int ci = 0; ci < cin; ++ci) {
                float xv;
                if (mode == 0)       xv = mm ? xraw[p] : 0.f;
                else if (mode == 1)  xv = (ci < 16) ? Xmain[p * 16 + ci]
                                                    : (mm ? xraw[p] : 0.f);
                else                 xv = Xmain[p * 16 + ci];
#pragma unroll
                for (int co = 0; co < 8; ++co) acc[co] += xv * wp[ci * 8 + co];
            }
        }
    }
    m1[tid] = any ? 1 : 0;
#pragma unroll
    for (int co = 0; co < 8; ++co)
        y1[(size_t)tid * 8 + co] = any ? fmaxf(acc[co] + b1[co], 0.f) : 0.f;
}

// ---------------------------------------------------------------------------
// k2: m2 = dilate3x3(m1 subsampled);  y2 = relu((conv3x3(y1_sub,w2)+b2)*m2)
// One thread per (b, oy, ox) on the H2 grid; 8->16 channels.
// ---------------------------------------------------------------------------
__global__ void k2_conv2(const float* __restrict__ y1,
                         const uint8_t* __restrict__ m1,
                         const float* __restrict__ w2,
                         const float* __restrict__ b2,
                         float* __restrict__ y2,
                         int s, int H1, int H2)
{
    int tid = blockIdx.x * blockDim.x + threadIdx.x;
    int total = 2 * H2 * H2;
    if (tid >= total) return;
    int b  = tid / (H2 * H2);
    int r  = tid - b * H2 * H2;
    int oy = r / H2, ox = r - (r / H2) * H2;

    float acc[16];
#pragma unroll
    for (int c = 0; c < 16; ++c) acc[c] = 0.f;
    bool any = false;

    for (int ky = 0; ky < 3; ++ky) {
        for (int kx = 0; kx < 3; ++kx) {
            int iy = s * (oy + ky), ix = s * (ox + kx);      // < H1 by construction
            size_t p = ((size_t)b * H1 + iy) * H1 + ix;
            any = any || (m1[p] != 0);
            const float* yv = y1 + p * 8;
            const float* wp = w2 + (size_t)(ky * 3 + kx) * 8 * 16;
#pragma unroll
            for (int ci = 0; ci < 8; ++ci) {
                float v = yv[ci];
#pragma unroll
                for (int co = 0; co < 16; ++co) acc[co] += v * wp[ci * 16 + co];
            }
        }
    }
#pragma unroll
    for (int co = 0; co < 16; ++co)
        y2[(size_t)tid * 16 + co] = any ? fmaxf(acc[co] + b2[co], 0.f) : 0.f;
}

// ---------------------------------------------------------------------------
// k3: y3 = relu((adjoint_conv(y2, w3) + b3) * m1) over the whole H1 grid.
// Coarse sites (q % s == 0 both dims) get the 16->8 full correlation of y2;
// everything else is just bias (masked).
// ---------------------------------------------------------------------------
__global__ void k3_inv1(const float* __restrict__ y2,
                        const uint8_t* __restrict__ m1,
                        const float* __restrict__ w3,
                        const float* __restrict__ b3,
                        float* __restrict__ y3,
                        int s, int H1, int H2)
{
    int tid = blockIdx.x * blockDim.x + threadIdx.x;
    int total = 2 * H1 * H1;
    if (tid >= total) return;
    int b  = tid / (H1 * H1);
    int r  = tid - b * H1 * H1;
    int qy = r / H1, qx = r - (r / H1) * H1;

    if (!m1[tid]) {
#pragma unroll
        for (int c = 0; c < 8; ++c) y3[(size_t)tid * 8 + c] = 0.f;
        return;
    }

    float acc[8];
#pragma unroll
    for (int c = 0; c < 8; ++c) acc[c] = 0.f;

    if ((qy % s) == 0 && (qx % s) == 0) {
        int iy = qy / s, ix = qx / s;
        for (int ky = 0; ky < 3; ++ky) {
            int sy = iy - ky;
            if (sy < 0 || sy >= H2) continue;
            for (int kx = 0; kx < 3; ++kx) {
                int sx = ix - kx;
                if (sx < 0 || sx >= H2) continue;
                const float* yv = y2 + (((size_t)b * H2 + sy) * H2 + sx) * 16;
                const float* wp = w3 + (size_t)(ky * 3 + kx) * 8 * 16;
#pragma unroll
                for (int ci = 0; ci < 8; ++ci) {
                    float a = 0.f;
#pragma unroll
                    for (int co = 0; co < 16; ++co) a += yv[co] * wp[ci * 16 + co];
                    acc[ci] += a;
                }
            }
        }
    }
#pragma unroll
    for (int ci = 0; ci < 8; ++ci)
        y3[(size_t)tid * 8 + ci] = fmaxf(acc[ci] + b3[ci], 0.f);
}

// ---------------------------------------------------------------------------
// k4: x_i = relu((adjoint_conv(y3, w4) + b4) * m0) on the full 601^2 grid.
// Fully writes the output tensor (zeros where m0 == 0), so later kernels may
// read it without any memset.
// ---------------------------------------------------------------------------
__global__ void k4_inv2(const float* __restrict__ y3,
                        const uint8_t* __restrict__ m0,
                        const float* __restrict__ w4,
                        const float* __restrict__ b4,
                        float* __restrict__ xout,
                        int s, int H1)
{
    int tid = blockIdx.x * blockDim.x + threadIdx.x;
    if (tid >= NPIX) return;
    int b  = tid / (H0 * H0);
    int r  = tid - b * H0 * H0;
    int py = r / H0, px = r - (r / H0) * H0;

    if (!m0[tid]) {
#pragma unroll
        for (int c = 0; c < 16; ++c) xout[(size_t)tid * 16 + c] = 0.f;
        return;
    }

    float acc[16];
#pragma unroll
    for (int c = 0; c < 16; ++c) acc[c] = 0.f;

    if ((py % s) == 0 && (px % s) == 0) {
        int qy = py / s, qx = px / s;                       // < Hs = H1+2
        for (int ky = 0; ky < 3; ++ky) {
            int sy = qy - ky;
            if (sy < 0 || sy >= H1) continue;
            for (int kx = 0; kx < 3; ++kx) {
                int sx = qx - kx;
                if (sx < 0 || sx >= H1) continue;
                const float* yv = y3 + (((size_t)b * H1 + sy) * H1 + sx) * 8;
                const float* wp = w4 + (size_t)(ky * 3 + kx) * 16 * 8;
#pragma unroll
                for (int ci = 0; ci < 16; ++ci) {
                    float a = 0.f;
#pragma unroll
                    for (int co = 0; co < 8; ++co) a += yv[co] * wp[ci * 8 + co];
                    acc[ci] += a;
                }
            }
        }
    }
#pragma unroll
    for (int ci = 0; ci < 16; ++ci)
        xout[(size_t)tid * 16 + ci] = fmaxf(acc[ci] + b4[ci], 0.f);
}

// ---------------------------------------------------------------------------
// k5: out[p, :] = m0[p] ? (concat(x1..x4)[p, :] @ wl + bl) : 0
// WMMA f32 16x16x4: one wave = 16 pixels (M) x 16 outputs (N), K = 64 swept
// in 16 chunks of 4.  A-fragment layout (ISA 7.12.2, 32-bit A 16x4):
//   lanes 0-15:  M=lane,    VGPR0/1 -> K = k0+0 / k0+1
//   lanes 16-31: M=lane-16, VGPR0/1 -> K = k0+2 / k0+3
// B (4x16): row-per-half-wave:  VGPR0 -> K = k0 + (lane>>4), N = lane&15;
//           VGPR1 -> K = k0+2+(lane>>4).
// C/D (16x16 f32): element (vgpr r, lane) = (M = r + 8*(lane>>4), N = lane&15).
// EXEC stays all-ones through the WMMAs (tail handled by clamped loads and
// guarded stores).
// ---------------------------------------------------------------------------
__global__ void k5_final_wmma(const float* __restrict__ x1f,
                              const float* __restrict__ x2f,
                              const float* __restrict__ x3f,
                              const float* __restrict__ x4f,
                              const uint8_t* __restrict__ m0,
                              const float* __restrict__ wl,
                              const float* __restrict__ bl,
                              float* __restrict__ out)
{
    const int lane = threadIdx.x & 31;
    const int wid  = (blockIdx.x * blockDim.x + threadIdx.x) >> 5;   // wave32
    const long base = (long)wid * 16;
    if (base >= NPIX) return;                                        // wave-uniform

    const float* bufs[4] = { x1f, x2f, x3f, x4f };
    const int m = lane & 15;        // pixel-in-tile for A, N for B/C/D
    const int g = lane >> 4;        // half-wave selector

    long pix = base + m;
    if (pix >= NPIX) pix = NPIX - 1;                                 // clamp tail loads

    // Prefetch next tile's A data into cache (global_prefetch_b8).
    if (base + 16 < NPIX) {
        __builtin_prefetch(bufs[0] + (base + 16 + m) * 16, 0, 0);
        __builtin_prefetch(bufs[2] + (base + 16 + m) * 16, 0, 0);
    }

    v8f c = {};
#pragma unroll
    for (int k0 = 0; k0 < 64; k0 += 4) {
        const float* xb = bufs[k0 >> 4] + pix * 16 + (k0 & 15) + g * 2;
        v2f a;  a.x = xb[0];              a.y = xb[1];
        const int kk = k0 + g;
        v2f bb; bb.x = wl[kk * 16 + m];   bb.y = wl[(kk + 2) * 16 + m];
        c = __builtin_amdgcn_wmma_f32_16x16x4_f32(
                /*neg_a=*/false, a, /*neg_b=*/false, bb,
                /*c_mod=*/(short)0, c, /*reuse_a=*/false, /*reuse_b=*/false);
    }

    const float bias = bl[m];
#pragma unroll
    for (int r8 = 0; r8 < 8; ++r8) {
        long prow = base + r8 + g * 8;                               // D-row -> pixel
        if (prow < NPIX) {
            float v = c[r8] + bias;
            out[prow * 16 + m] = m0[prow] ? v : 0.f;
        }
    }
}

// ---------------------------------------------------------------------------
// Host launch.  d_in pytree order (sorted-key flatten of params):
//   0: x (2*601*601*1 f32)   1: mask (2*601*601 bool/uint8)
//   2: bl(16)
//   3+8i .. 10+8i for block i: b1(8) b2(16) b3(8) b4(16) w1 w2 w3 w4
//   35: wl (64*16)
// ---------------------------------------------------------------------------
extern "C" void kernel_launch(void* const* d_in, const int* in_sizes, int n_in,
                              void* d_out, int out_size, void* d_ws, size_t ws_size,
                              hipStream_t stream) {
    (void)in_sizes; (void)n_in; (void)out_size; (void)ws_size;

    const float*   x    = (const float*)d_in[0];
    const uint8_t* mask = (const uint8_t*)d_in[1];   // numpy bool: 1 byte/elem
    const float*   bl   = (const float*)d_in[2];
    const float*   wl   = (const float*)d_in[35];

    // Block geometry (stride == dilation == s).
    const int S[4]   = {4, 3, 2, 1};
    const int H1a[4] = {149, 199, 299, 599};   // conv1 output grid
    const int H2a[4] = {36, 65, 148, 597};     // conv2 output grid
    const int CIN[4] = {1, 17, 16, 16};
    const int MODE[4]= {0, 1, 2, 2};

    // Workspace carve-up.
    char* ws = (char*)d_ws;
    const size_t FULL = (size_t)NPIX * 16 * sizeof(float);   // 46,233,728 B
    float* xf[4];
    for (int i = 0; i < 4; ++i) xf[i] = (float*)(ws + (size_t)i * FULL);
    char* scr = ws + 4 * FULL;
    float*   y1 = (float*)scr;                                scr += (size_t)2 * 599 * 599 * 8  * sizeof(float);
    float*   y2 = (float*)scr;                                scr += (size_t)2 * 597 * 597 * 16 * sizeof(float);
    float*   y3 = (float*)scr;                                scr += (size_t)2 * 599 * 599 * 8  * sizeof(float);
    uint8_t* m1 = (uint8_t*)scr;

    const int TPB = 256;
    for (int i = 0; i < 4; ++i) {
        const int s = S[i], H1 = H1a[i], H2 = H2a[i];
        const float* b1 = (const float*)d_in[3 + 8 * i + 0];
        const float* b2 = (const float*)d_in[3 + 8 * i + 1];
        const float* b3 = (const float*)d_in[3 + 8 * i + 2];
        const float* b4 = (const float*)d_in[3 + 8 * i + 3];
        const float* w1 = (const float*)d_in[3 + 8 * i + 4];
        const float* w2 = (const float*)d_in[3 + 8 * i + 5];
        const float* w3 = (const float*)d_in[3 + 8 * i + 6];
        const float* w4 = (const float*)d_in[3 + 8 * i + 7];
        const float* Xmain = (i == 0) ? nullptr : xf[i - 1];

        int n1 = 2 * H1 * H1;
        hipLaunchKernelGGL(k1_conv1, dim3((n1 + TPB - 1) / TPB), dim3(TPB), 0, stream,
                           Xmain, x, mask, w1, b1, y1, m1, s, H1, CIN[i], MODE[i]);
        int n2 = 2 * H2 * H2;
        hipLaunchKernelGGL(k2_conv2, dim3((n2 + TPB - 1) / TPB), dim3(TPB), 0, stream,
                           y1, m1, w2, b2, y2, s, H1, H2);
        hipLaunchKernelGGL(k3_inv1, dim3((n1 + TPB - 1) / TPB), dim3(TPB), 0, stream,
                           y2, m1, w3, b3, y3, s, H1, H2);
        hipLaunchKernelGGL(k4_inv2, dim3((NPIX + TPB - 1) / TPB), dim3(TPB), 0, stream,
                           y3, mask, w4, b4, xf[i], s, H1);
    }

    // Final fused 64->16 matmul + bias + mask on the matrix pipe.
    const long nwaves  = (NPIX + 15) / 16;            // 45,151 tiles
    const long threads = nwaves * 32;
    hipLaunchKernelGGL(k5_final_wmma, dim3((threads + TPB - 1) / TPB), dim3(TPB), 0, stream,
                       xf[0], xf[1], xf[2], xf[3], mask, wl, bl, (float*)d_out);
}